// BoundsReLUWrapper_70652212019204
// MI455X (gfx1250) — compile-verified
//
#include <hip/hip_runtime.h>
#include <math.h>

// Problem dims (fixed by setup_inputs: B=1, L=256, DI=256, DO=256, p=2)
#define L_DIM   256
#define DI_DIM  256
#define DO_DIM  256
#define W_ELEMS (L_DIM * DI_DIM * DO_DIM)   // 16,777,216 per weight tensor
#define COLS    (L_DIM * DO_DIM)            // 65,536 (l, do) columns
#define OFF_LB  (W_ELEMS)                   // d_out layout: [lw_o][lb_o][uw_o][ub_o]
#define OFF_UW  (W_ELEMS + COLS)
#define OFF_UB  (W_ELEMS + COLS + W_ELEMS)
#define EPS_REF 1e-12f

typedef float v2f __attribute__((ext_vector_type(2)));
typedef float v8f __attribute__((ext_vector_type(8)));

#if __has_builtin(__builtin_amdgcn_wmma_f32_16x16x4_f32)
#define HAVE_WMMA_F32_K4 1
#else
#define HAVE_WMMA_F32_K4 0
#endif

// Per-column epilogue: given sum-of-squares for lw/uw columns, derive the
// ReLU relaxation scale factors and output biases.
__device__ __forceinline__ void column_epilogue(
    float sumL, float sumU, float lb_c, float ub_c, float eps_s,
    float& s_l, float& s_u, float& lb_o, float& ub_o)
{
    const float lv = lb_c - eps_s * sqrtf(sumL);
    const float uv = ub_c + eps_s * sqrtf(sumU);
    if (lv > 0.0f) {                 // strictly positive: identity
        s_l = 1.0f; s_u = 1.0f; lb_o = lb_c; ub_o = ub_c;
    } else if (uv < 0.0f) {          // strictly negative: zero
        s_l = 0.0f; s_u = 0.0f; lb_o = 0.0f; ub_o = 0.0f;
    } else {                         // crossing zero
        const float ku = uv / (uv - lv + EPS_REF);      // k_u >= 0 here
        s_u  = ku;
        ub_o = (ub_c - lv) * ku;
        const float kl = (fabsf(uv) > fabsf(lv)) ? 1.0f : 0.0f;
        s_l  = kl;
        lb_o = kl * lb_c;
    }
}

#if HAVE_WMMA_F32_K4
// ---------------------------------------------------------------------------
// Pass 1 (WMMA): column sum-of-squares over DI via V_WMMA_F32_16X16X4_F32.
//   A (16x4, M=do, K=di) = squared weights, B (4x16) = ones
//   => D[m,n] = sum_k A[m,k] for every n; accumulate 64 K=4 steps -> full sum.
// Block = 256 threads = 8 waves; each wave owns 16 consecutive `do` columns;
// 2 blocks per l row (grid = 2*L).
// A layout (ISA 7.12.2, 32-bit A 16x4): lanes 0-15 -> K={0,1} in v[0],v[1];
// lanes 16-31 -> K={2,3}. C/D layout: lane 0 holds M=0..7, lane 16 M=8..15.
// ---------------------------------------------------------------------------
__global__ __launch_bounds__(256) void colnorm_scales_wmma(
    const float* __restrict__ lw, const float* __restrict__ lb,
    const float* __restrict__ uw, const float* __restrict__ ub,
    const float* __restrict__ eps_p,
    float* __restrict__ out, float* __restrict__ sl, float* __restrict__ su)
{
    const int l       = blockIdx.x >> 1;
    const int half    = blockIdx.x & 1;
    const int wave    = threadIdx.x >> 5;
    const int lane    = threadIdx.x & 31;
    const int do_base = half * 128 + wave * 16;
    const int m       = lane & 15;       // M index (do offset)
    const int kh      = lane >> 4;       // 0 -> K{0,1}, 1 -> K{2,3}

    const size_t base = (size_t)l * (DI_DIM * DO_DIM)
                      + (size_t)(2 * kh) * DO_DIM + do_base + m;
    const float* lp = lw + base;
    const float* up = uw + base;

    v8f accL = {0.f,0.f,0.f,0.f,0.f,0.f,0.f,0.f};
    v8f accU = {0.f,0.f,0.f,0.f,0.f,0.f,0.f,0.f};
    const v2f ones = {1.0f, 1.0f};

    for (int di0 = 0; di0 < DI_DIM; di0 += 4) {
        const float a0 = lp[0], a1 = lp[DO_DIM];
        const float b0 = up[0], b1 = up[DO_DIM];
        const v2f aL = { a0 * a0, a1 * a1 };
        const v2f aU = { b0 * b0, b1 * b1 };
        accL = __builtin_amdgcn_wmma_f32_16x16x4_f32(
                   false, aL, false, ones, (short)0, accL, false, false);
        accU = __builtin_amdgcn_wmma_f32_16x16x4_f32(
                   false, aU, false, ones, (short)0, accU, false, false);
        lp += 4 * DO_DIM;
        up += 4 * DO_DIM;
    }

    // Writer lanes: lane 0 holds M=0..7, lane 16 holds M=8..15 (any N column;
    // all N are identical since B is all-ones).
    if ((lane & 15) == 0) {
        const float eps_s = eps_p[0];
        const int mo = (lane >> 4) * 8;
        #pragma unroll
        for (int r = 0; r < 8; ++r) {
            const int d   = do_base + mo + r;
            const int col = l * DO_DIM + d;
            float s_l, s_u, lb_o, ub_o;
            column_epilogue(accL[r], accU[r], lb[col], ub[col], eps_s,
                            s_l, s_u, lb_o, ub_o);
            sl[col] = s_l;
            su[col] = s_u;
            out[(size_t)OFF_LB + col] = lb_o;
            out[(size_t)OFF_UB + col] = ub_o;
        }
    }
}
#else
// Scalar fallback: one thread per column, coalesced across `do`.
__global__ __launch_bounds__(256) void colnorm_scales_scalar(
    const float* __restrict__ lw, const float* __restrict__ lb,
    const float* __restrict__ uw, const float* __restrict__ ub,
    const float* __restrict__ eps_p,
    float* __restrict__ out, float* __restrict__ sl, float* __restrict__ su)
{
    const int l = blockIdx.x;
    const int d = threadIdx.x;
    const size_t base = (size_t)l * (DI_DIM * DO_DIM) + d;
    float sumL = 0.f, sumU = 0.f;
    for (int di = 0; di < DI_DIM; ++di) {
        const float a = lw[base + (size_t)di * DO_DIM];
        const float b = uw[base + (size_t)di * DO_DIM];
        sumL = fmaf(a, a, sumL);
        sumU = fmaf(b, b, sumU);
    }
    const int col = l * DO_DIM + d;
    float s_l, s_u, lb_o, ub_o;
    column_epilogue(sumL, sumU, lb[col], ub[col], eps_p[0], s_l, s_u, lb_o, ub_o);
    sl[col] = s_l;
    su[col] = s_u;
    out[(size_t)OFF_LB + col] = lb_o;
    out[(size_t)OFF_UB + col] = ub_o;
}
#endif

// ---------------------------------------------------------------------------
// Pass 2: pure streaming scale. float4 (b128) loads/stores; scales fetched as
// float4 (contiguous in `do`). Re-reads of lw/uw should hit the 192MB L2.
// ---------------------------------------------------------------------------
__global__ __launch_bounds__(256) void scale_weights(
    const float4* __restrict__ lw4, const float4* __restrict__ uw4,
    const float4* __restrict__ sl4, const float4* __restrict__ su4,
    float4* __restrict__ out_lw, float4* __restrict__ out_uw)
{
    const int i4 = blockIdx.x * blockDim.x + threadIdx.x;   // < W_ELEMS/4
    const int i  = i4 << 2;
    const int l  = i >> 16;          // / (DI*DO)
    const int d  = i & (DO_DIM - 1); // do of first component (multiple of 4)
    const int s  = (l << 6) | (d >> 2);

    const float4 s_l = sl4[s];
    const float4 s_u = su4[s];
    const float4 a   = lw4[i4];
    const float4 b   = uw4[i4];

    float4 ol, ou;
    ol.x = a.x * s_l.x; ol.y = a.y * s_l.y; ol.z = a.z * s_l.z; ol.w = a.w * s_l.w;
    ou.x = b.x * s_u.x; ou.y = b.y * s_u.y; ou.z = b.z * s_u.z; ou.w = b.w * s_u.w;

    out_lw[i4] = ol;
    out_uw[i4] = ou;
}

extern "C" void kernel_launch(void* const* d_in, const int* in_sizes, int n_in,
                              void* d_out, int out_size, void* d_ws, size_t ws_size,
                              hipStream_t stream) {
    // inputs: 0=p (int, ==2 per harness), 1=eps (f32[1]), 2=lw, 3=lb, 4=uw, 5=ub
    const float* eps = (const float*)d_in[1];
    const float* lw  = (const float*)d_in[2];
    const float* lb  = (const float*)d_in[3];
    const float* uw  = (const float*)d_in[4];
    const float* ub  = (const float*)d_in[5];
    float* out = (float*)d_out;

    float* sl = (float*)d_ws;          // COLS floats
    float* su = sl + COLS;             // COLS floats  (512 KB total)

#if HAVE_WMMA_F32_K4
    colnorm_scales_wmma<<<2 * L_DIM, 256, 0, stream>>>(lw, lb, uw, ub, eps,
                                                       out, sl, su);
#else
    colnorm_scales_scalar<<<L_DIM, 256, 0, stream>>>(lw, lb, uw, ub, eps,
                                                     out, sl, su);
#endif

    scale_weights<<<W_ELEMS / 4 / 256, 256, 0, stream>>>(
        (const float4*)lw, (const float4*)uw,
        (const float4*)sl, (const float4*)su,
        (float4*)out, (float4*)(out + OFF_UW));
}